// GPUSpectralEncoder_82703890252548
// MI455X (gfx1250) — compile-verified
//
#include <hip/hip_runtime.h>
#include <hip/hip_bf16.h>
#include <math.h>

// ---------------------------------------------------------------------------
// Spectral encoder for MI455X (gfx1250, wave32).
//
// Dense B = SIGMA*I - L (64MB, L2-resident in the 192MB L2); block power
// iteration with a 32-vector subspace; inner loop is a FP32 WMMA GEMM
// (v_wmma_f32_16x16x4_f32) where each wave computes a 16x32 output strip so
// every element of B is loaded exactly once per iteration (L2-traffic-optimal).
// Rayleigh-Ritz (32x32 Jacobi) -> eigenvalues [1..16] -> tiny MLP.
//
// Workspace requirement: ~70.3 MB.
// ---------------------------------------------------------------------------

#define NNODES 4096
#define NWORDS 128              // 4096 bits / 32 per row
#define NBLK   32               // subspace block width
#define SIGMA  2.01f            // > max eig of normalized Laplacian (<= 2 + 1e-6)
#define NITER  96

typedef __attribute__((ext_vector_type(2))) float v2f;
typedef __attribute__((ext_vector_type(8))) float v8f;

// ---------------- graph construction ----------------

__global__ void k_zero_u32(unsigned int* p, int n) {
  int i = blockIdx.x * blockDim.x + threadIdx.x;
  if (i < n) p[i] = 0u;
}

__global__ void k_scatter_edges(const int* __restrict__ ei, int E,
                                unsigned int* __restrict__ bits) {
  int e = blockIdx.x * blockDim.x + threadIdx.x;
  if (e >= E) return;
  int r = ei[e];
  int c = ei[E + e];
  atomicOr(&bits[(size_t)r * NWORDS + (c >> 5)], 1u << (c & 31));
  atomicOr(&bits[(size_t)c * NWORDS + (r >> 5)], 1u << (r & 31));
}

__global__ void k_degrees(const unsigned int* __restrict__ bits,
                          float* __restrict__ dis) {
  int i = blockIdx.x * blockDim.x + threadIdx.x;
  if (i >= NNODES) return;
  int d = 0;
  for (int w = 0; w < NWORDS; ++w) d += __popc(bits[(size_t)i * NWORDS + w]);
  dis[i] = (d > 0) ? rsqrtf((float)d) : 0.0f;
}

// B[i][j] = (i==j ? SIGMA - 1 - 1e-6 : 0) + A[i][j]*dis[i]*dis[j]
__global__ __launch_bounds__(256) void k_build_B(const unsigned int* __restrict__ bits,
                                                 const float* __restrict__ dis,
                                                 float* __restrict__ Bm) {
  size_t idx = (size_t)blockIdx.x * 256 + threadIdx.x;   // 16.7M elements
  int i = (int)(idx >> 12);
  int j = (int)(idx & 4095);
  unsigned int w = bits[(size_t)i * NWORDS + (j >> 5)];
  float a = ((w >> (j & 31)) & 1u) ? dis[i] * dis[j] : 0.0f;
  float v = a + ((i == j) ? (SIGMA - 1.0f - 1e-6f) : 0.0f);
  Bm[idx] = v;
}

__global__ void k_init_X(float* __restrict__ X) {
  int idx = blockIdx.x * blockDim.x + threadIdx.x;       // 4096*32
  unsigned int h = (unsigned int)idx * 2654435761u;
  h ^= h >> 16; h *= 2246822519u; h ^= h >> 13;
  X[idx] = ((float)(h & 0xFFFFu) / 65536.0f) - 0.5f;
}

// ---------------- WMMA FP32 GEMM: Y = B (4096x4096) * X (4096x32) ----------------
// One wave per 16x32 output strip (two 16x16 WMMA tiles sharing the A fragment,
// so B-matrix elements are loaded exactly once per GEMM).
// A operand 16x4 f32 (2 VGPRs): lanes 0-15 hold {K=k,k+1} of row M=lane;
//   lanes 16-31 hold {K=k+2,k+3}.
// B operand 4x16 f32 (2 VGPRs): lanes 0-15 hold {K=k,k+1} of col N=lane;
//   lanes 16-31 hold {K=k+2,k+3}.
// C/D 16x16 f32: VGPR v -> row v (lanes 0-15) / row v+8 (lanes 16-31).
__global__ __launch_bounds__(128) void k_gemm_wmma(const float* __restrict__ Bm,
                                                   const float* __restrict__ X,
                                                   float* __restrict__ Y) {
  const int N = NNODES;
  int tile_m = blockIdx.x * (blockDim.x >> 5) + (threadIdx.x >> 5);  // 0..255
  int lane = threadIdx.x & 31;
  int half = lane >> 4;
  int lid  = lane & 15;

  int arow = (tile_m << 4) + lid;               // A row this lane serves
  const float* ap = Bm + (size_t)arow * N + (half << 1);
  const float* bp = X + (size_t)(half << 1) * NBLK + lid;

  v8f acc0 = {};
  v8f acc1 = {};
  for (int k = 0; k < N; k += 4) {
    __builtin_prefetch(ap + k + 512, 0, 3);                 // keep-near prefetch of A stream
    v2f a = *reinterpret_cast<const v2f*>(ap + k);          // {B[r][k+2h], B[r][k+2h+1]}
    v2f b0, b1;
    b0.x = bp[(size_t)k * NBLK];                            // X[k+2h  ][lid]
    b0.y = bp[(size_t)k * NBLK + NBLK];                     // X[k+2h+1][lid]
    b1.x = bp[(size_t)k * NBLK + 16];                       // X[k+2h  ][16+lid]
    b1.y = bp[(size_t)k * NBLK + NBLK + 16];                // X[k+2h+1][16+lid]
    acc0 = __builtin_amdgcn_wmma_f32_16x16x4_f32(
        false, a, false, b0, (short)0, acc0, false, false);
    acc1 = __builtin_amdgcn_wmma_f32_16x16x4_f32(
        false, a, false, b1, (short)0, acc1, false, false);
  }

  int rbase = (tile_m << 4) + (half << 3);
#pragma unroll
  for (int v = 0; v < 8; ++v) {
    Y[(size_t)(rbase + v) * NBLK + lid]      = acc0[v];
    Y[(size_t)(rbase + v) * NBLK + 16 + lid] = acc1[v];
  }
}

// ---------------- small dense helpers ----------------

// G[a][c] = sum_i A[i][a] * Bv[i][c]   (grid = 1024 blocks, one entry each)
__global__ __launch_bounds__(128) void k_gram2(const float* __restrict__ A,
                                               const float* __restrict__ Bv,
                                               float* __restrict__ G) {
  int a = blockIdx.x >> 5;
  int c = blockIdx.x & 31;
  __shared__ float red[128];
  float s = 0.0f;
  for (int i = threadIdx.x; i < NNODES; i += 128)
    s += A[(size_t)i * NBLK + a] * Bv[(size_t)i * NBLK + c];
  red[threadIdx.x] = s;
  __syncthreads();
  for (int off = 64; off > 0; off >>= 1) {
    if (threadIdx.x < off) red[threadIdx.x] += red[threadIdx.x + off];
    __syncthreads();
  }
  if (threadIdx.x == 0) G[a * NBLK + c] = red[0];
}

// Upper Cholesky G = R^T R, then Rinv = R^{-1} (upper). Single thread; 32x32.
__global__ void k_cholinv(const float* __restrict__ G, float* __restrict__ Rinv) {
  if (threadIdx.x != 0 || blockIdx.x != 0) return;
  __shared__ float R[NBLK * NBLK];
  __shared__ float Ri[NBLK * NBLK];
  for (int j = 0; j < NBLK; ++j) {
    for (int i = 0; i <= j; ++i) {
      float s = G[i * NBLK + j];
      for (int k = 0; k < i; ++k) s -= R[k * NBLK + i] * R[k * NBLK + j];
      if (i == j) R[j * NBLK + j] = sqrtf(fmaxf(s, 1e-30f));
      else        R[i * NBLK + j] = s / R[i * NBLK + i];
    }
  }
  for (int i = 0; i < NBLK; ++i)
    for (int j = 0; j < NBLK; ++j) Ri[i * NBLK + j] = 0.0f;
  for (int i = 0; i < NBLK; ++i) {
    Ri[i * NBLK + i] = 1.0f / R[i * NBLK + i];
    for (int j = i + 1; j < NBLK; ++j) {
      float s = 0.0f;
      for (int k = i; k < j; ++k) s += Ri[i * NBLK + k] * R[k * NBLK + j];
      Ri[i * NBLK + j] = -s / R[j * NBLK + j];
    }
  }
  for (int i = 0; i < NBLK * NBLK; ++i) Rinv[i] = Ri[i];
}

// Xout = Xin * Rinv  (Rinv upper triangular)
__global__ __launch_bounds__(256) void k_applyR(const float* __restrict__ Xin,
                                                const float* __restrict__ Rinv,
                                                float* __restrict__ Xout) {
  int idx = blockIdx.x * 256 + threadIdx.x;   // 4096*32
  int i = idx >> 5;
  int j = idx & 31;
  float s = 0.0f;
  for (int k = 0; k <= j; ++k) s += Xin[(size_t)i * NBLK + k] * Rinv[k * NBLK + j];
  Xout[idx] = s;
}

// Jacobi eigensolve of M = SIGMA*I - sym(S) (32x32); spec = sorted eigs [1..16].
__global__ void k_jacobi_spec(const float* __restrict__ S, float* __restrict__ spec) {
  __shared__ float M[NBLK][NBLK];
  __shared__ float cs[2];
  int t = threadIdx.x;
  for (int j = 0; j < NBLK; ++j)
    M[t][j] = ((t == j) ? SIGMA : 0.0f) - 0.5f * (S[t * NBLK + j] + S[j * NBLK + t]);
  __syncthreads();
  for (int sweep = 0; sweep < 16; ++sweep) {
    for (int p = 0; p < NBLK - 1; ++p) {
      for (int q = p + 1; q < NBLK; ++q) {
        if (t == 0) {
          float apq = M[p][q];
          float c = 1.0f, s = 0.0f;
          if (fabsf(apq) > 1e-12f) {
            float tau = (M[q][q] - M[p][p]) / (2.0f * apq);
            float tt = ((tau >= 0.0f) ? 1.0f : -1.0f) / (fabsf(tau) + sqrtf(1.0f + tau * tau));
            c = 1.0f / sqrtf(1.0f + tt * tt);
            s = tt * c;
          }
          cs[0] = c; cs[1] = s;
        }
        __syncthreads();
        float c = cs[0], s = cs[1];
        float mp = M[p][t], mq = M[q][t];          // row rotation
        M[p][t] = c * mp - s * mq;
        M[q][t] = s * mp + c * mq;
        __syncthreads();
        mp = M[t][p]; mq = M[t][q];                // column rotation
        M[t][p] = c * mp - s * mq;
        M[t][q] = s * mp + c * mq;
        __syncthreads();
      }
    }
  }
  if (t == 0) {
    float ev[NBLK];
    for (int i = 0; i < NBLK; ++i) ev[i] = M[i][i];
    for (int i = 1; i < NBLK; ++i) {               // insertion sort ascending
      float v = ev[i]; int j = i - 1;
      while (j >= 0 && ev[j] > v) { ev[j + 1] = ev[j]; --j; }
      ev[j + 1] = v;
    }
    for (int k = 0; k < 16; ++k) spec[k] = ev[k + 1];   // skip trivial eig
  }
}

// out = relu(spec @ W1 + b1) @ W2 + b2
__global__ __launch_bounds__(128) void k_mlp(const float* __restrict__ spec,
                                             const float* __restrict__ W1,
                                             const float* __restrict__ b1,
                                             const float* __restrict__ W2,
                                             const float* __restrict__ b2,
                                             float* __restrict__ out) {
  __shared__ float h[128];
  int j = threadIdx.x;
  float s = b1[j];
  for (int k = 0; k < 16; ++k) s += spec[k] * W1[k * 128 + j];
  h[j] = fmaxf(s, 0.0f);
  __syncthreads();
  float o = b2[j];
  for (int k = 0; k < 128; ++k) o += h[k] * W2[k * 128 + j];
  out[j] = o;
}

// ---------------- host orchestration ----------------

extern "C" void kernel_launch(void* const* d_in, const int* in_sizes, int n_in,
                              void* d_out, int out_size, void* d_ws, size_t ws_size,
                              hipStream_t stream) {
  const int* ei    = (const int*)d_in[0];
  const float* W1  = (const float*)d_in[2];
  const float* b1  = (const float*)d_in[3];
  const float* W2  = (const float*)d_in[4];
  const float* b2  = (const float*)d_in[5];
  float* out = (float*)d_out;
  const int E = in_sizes[0] / 2;

  char* ws = (char*)d_ws;
  // Workspace layout (256B aligned offsets), total ~70.3 MB
  float*        Bm   = (float*)(ws + 0);                      // 64 MB dense B = SIGMA*I - L
  unsigned int* bits = (unsigned int*)(ws + 67108864);        // 2 MB adjacency bitset
  float*        dis  = (float*)(ws + 69206016);               // 16 KB  D^{-1/2}
  float*        X    = (float*)(ws + 69222400);               // 512 KB block vectors
  float*        Yb   = (float*)(ws + 69746688);               // 512 KB block vectors
  float*        G    = (float*)(ws + 70270976);               // 4 KB Gram
  float*        Ri   = (float*)(ws + 70275072);               // 4 KB R^{-1}
  float*        Sm   = (float*)(ws + 70279168);               // 4 KB Ritz matrix
  float*        spec = (float*)(ws + 70283264);               // 64 B eigenvalues 1..16

  const int NBITW = NNODES * NWORDS;   // 524288 words

  // 1) adjacency bitset (exact dedupe, matches reference set-semantics)
  k_zero_u32<<<(NBITW + 255) / 256, 256, 0, stream>>>(bits, NBITW);
  k_scatter_edges<<<(E + 255) / 256, 256, 0, stream>>>(ei, E, bits);
  k_degrees<<<NNODES / 128, 128, 0, stream>>>(bits, dis);

  // 2) dense B = SIGMA*I - L (64 MB -> L2 resident thereafter)
  k_build_B<<<(NNODES * NNODES) / 256, 256, 0, stream>>>(bits, dis, Bm);

  // 3) subspace iteration: orthonormalize, then X <- B * Q (WMMA FP32 GEMM)
  k_init_X<<<(NNODES * NBLK) / 256, 256, 0, stream>>>(X);
  for (int it = 0; it < NITER; ++it) {
    k_gram2<<<NBLK * NBLK, 128, 0, stream>>>(X, X, G);
    k_cholinv<<<1, 1, 0, stream>>>(G, Ri);
    k_applyR<<<(NNODES * NBLK) / 256, 256, 0, stream>>>(X, Ri, Yb);
    k_gemm_wmma<<<64, 128, 0, stream>>>(Bm, Yb, X);
  }

  // 4) Rayleigh-Ritz: Q = orth(X); S = Q^T B Q; eig(SIGMA*I - S) -> spec[1..16]
  k_gram2<<<NBLK * NBLK, 128, 0, stream>>>(X, X, G);
  k_cholinv<<<1, 1, 0, stream>>>(G, Ri);
  k_applyR<<<(NNODES * NBLK) / 256, 256, 0, stream>>>(X, Ri, Yb);   // Yb = Q
  k_gemm_wmma<<<64, 128, 0, stream>>>(Bm, Yb, X);                   // X = B Q
  k_gram2<<<NBLK * NBLK, 128, 0, stream>>>(Yb, X, Sm);              // S = Q^T B Q
  k_jacobi_spec<<<1, 32, 0, stream>>>(Sm, spec);

  // 5) MLP head
  k_mlp<<<1, 128, 0, stream>>>(spec, W1, b1, W2, b2, out);

  (void)n_in; (void)out_size; (void)ws_size;
}